// RelativePositionEncoding_53300544143969
// MI455X (gfx1250) — compile-verified
//
#include <hip/hip_runtime.h>
#include <stdint.h>

// ---------------------------------------------------------------------------
// RelativePositionEncoding for MI455X (gfx1250, wave32).
//
// out[i,j,0:128] = W_r[d_res(i,j)] + W_t[d_tok(i,j)] + same_entity(i,j)*w_e
//                + W_s[d_chain(i,j)]
//
// Store-bandwidth-bound: 1.21 GB write-only output -> ~52 us floor @ 23.3 TB/s.
// CDNA5 paths: global_load_async_to_lds_b128 (ASYNCcnt) DMAs the 71 KB W table
// into LDS with 18 straight-line issues using the dual-applied IOFFSET;
// s_wait_asynccnt 0 split-counter wait; ds_load_b128 gathers; 128-bit NT
// global stores so the output stream doesn't thrash the 192 MB L2.
// ---------------------------------------------------------------------------

typedef float f32x4 __attribute__((ext_vector_type(4)));
typedef int   i32x4 __attribute__((ext_vector_type(4)));

#define R_MAX 32
#define S_MAX 2
#define C_Z   128
#define NQ    (C_Z / 4)                 // 32 float4 per table row
#define N_R   (2 * R_MAX + 2)           // 66
#define N_S   (2 * S_MAX + 2)           // 6
#define D_IN  (N_R + N_R + 1 + N_S)     // 139
#define TBL4  (D_IN * NQ)               // 4448 float4 = 71168 B
#define MAXN  1536

__device__ __forceinline__ int clamp_i(int v, int hi) {
    v = v < 0 ? 0 : v;
    return v > hi ? hi : v;
}

// One async 16B/lane copy; IOFFSET is added to BOTH the LDS and global address.
#define ALD(OFF)                                                               \
    asm volatile("global_load_async_to_lds_b128 %0, %1, off offset:" #OFF     \
                 :: "v"(loff), "v"(gaddr) : "memory")

__global__ __launch_bounds__(256) void relpos_kernel(
    const int* __restrict__ asym, const int* __restrict__ res,
    const int* __restrict__ ent,  const int* __restrict__ tok,
    const int* __restrict__ sym,  const float* __restrict__ W,
    float* __restrict__ out, int N)
{
    __shared__ f32x4 tbl[TBL4];     // 71168 B: the whole W table
    __shared__ i32x4 desc[MAXN];    // 24576 B: per-j packed gather descriptors

    const int tid = threadIdx.x;
    const int i   = blockIdx.x;

    // ---- Phase 1a: async DMA of W (139 x 128 f32) into LDS ------------------
    // 256 lanes x 16 B x 18 issues covers 4448 float4 (tail: 96 lanes).
    {
        const uint32_t           loff  = (uint32_t)(uintptr_t)(&tbl[tid]);
        const unsigned long long gaddr =
            (unsigned long long)(uintptr_t)((const f32x4*)W + tid);
        ALD(0);     ALD(4096);  ALD(8192);  ALD(12288);
        ALD(16384); ALD(20480); ALD(24576); ALD(28672);
        ALD(32768); ALD(36864); ALD(40960); ALD(45056);
        ALD(49152); ALD(53248); ALD(57344); ALD(61440);
        ALD(65536);                       // 17 full rounds = 4352 float4
        if (tid < (TBL4 - 17 * 256)) {    // tail: 96 float4
            ALD(69632);
        }
    }

    // ---- Phase 1b (overlapped with the async DMA): per-j descriptors --------
    const int ai = asym[i], ri = res[i], ei = ent[i], ti = tok[i], si = sym[i];
    for (int j = tid; j < N; j += 256) {
        const int aj = asym[j], rj = res[j], ej = ent[j], tj = tok[j], sj = sym[j];
        const bool sc = (ai == aj);

        int dr = clamp_i(ri - rj + R_MAX, 2 * R_MAX);
        dr = sc ? dr : (2 * R_MAX + 1);

        int dt = clamp_i(ti - tj + R_MAX, 2 * R_MAX);
        dt = (sc && (ri == rj)) ? dt : (2 * R_MAX + 1);

        int dc = clamp_i(si - sj + S_MAX, 2 * S_MAX);
        dc = sc ? dc : (2 * S_MAX + 1);

        const float e = (ei == ej) ? 1.0f : 0.0f;

        i32x4 d;
        d.x = dr * NQ;                     // float4 index of W_r row
        d.y = (N_R + dt) * NQ;             // float4 index of W_t row
        d.z = (2 * N_R + 1 + dc) * NQ;     // float4 index of W_s row
        d.w = __float_as_int(e);
        desc[j] = d;
    }

    asm volatile("s_wait_asynccnt 0" ::: "memory");  // my async->LDS complete
    __syncthreads();                                 // all waves' LDS visible

    // ---- Phase 2: stream output rows. wave w owns j = w, w+8, w+16, ... -----
    const int lane = tid & 31;
    const int wave = tid >> 5;

    const f32x4 we = tbl[(2 * N_R) * NQ + lane];     // w_e row lives in regs
    f32x4* __restrict__ op =
        (f32x4*)out + ((size_t)i * N + wave) * NQ + lane;

    #pragma unroll 4
    for (int j = wave; j < N; j += 8, op += 8 * NQ) {
        const i32x4 d = desc[j];                     // ds_load_b128 broadcast
        const f32x4 r = tbl[d.x + lane];             // ds_load_b128 gathers
        const f32x4 t = tbl[d.y + lane];
        const f32x4 s = tbl[d.z + lane];
        const float e = __int_as_float(d.w);

        f32x4 o;
        o.x = fmaf(e, we.x, r.x + t.x + s.x);
        o.y = fmaf(e, we.y, r.y + t.y + s.y);
        o.z = fmaf(e, we.z, r.z + t.z + s.z);
        o.w = fmaf(e, we.w, r.w + t.w + s.w);

        __builtin_nontemporal_store(o, op);          // 128-bit NT store
    }
}

extern "C" void kernel_launch(void* const* d_in, const int* in_sizes, int n_in,
                              void* d_out, int out_size, void* d_ws, size_t ws_size,
                              hipStream_t stream) {
    const int*   asym = (const int*)  d_in[0];
    const int*   res  = (const int*)  d_in[1];
    const int*   ent  = (const int*)  d_in[2];
    const int*   tok  = (const int*)  d_in[3];
    const int*   sym  = (const int*)  d_in[4];
    const float* W    = (const float*)d_in[5];
    float*       out  = (float*)d_out;

    const int N = in_sizes[0];   // B == 1, N = 1536 (<= MAXN)

    relpos_kernel<<<dim3(N), dim3(256), 0, stream>>>(asym, res, ent, tok, sym,
                                                     W, out, N);
}